// HashGridEncoder_68925635166659
// MI455X (gfx1250) — compile-verified
//
#include <hip/hip_runtime.h>

// Instant-NGP style multiresolution hash-grid encoder for MI455X (gfx1250).
//
// Memory-bound gather workload: 16M (point,level) work items, each doing
// 8x 8-byte hashed gathers from a 64MB table (fits entirely in the 192MB L2)
// and a trilinear blend. No matmul structure -> WMMA not applicable; the win
// is memory-path shaping: temporal gathers (table stays in L2), non-temporal
// streaming loads/stores for x/out (don't evict the table), gfx1250
// global_prefetch for the x stream, and a wave32-friendly layout that makes
// the output store one contiguous 256B NT store per wave.

#define LVLS 16
#define LOG2_T 19
#define TBL (1u << LOG2_T)
#define HMASK (TBL - 1u)

typedef __attribute__((ext_vector_type(2))) float vf2;

// floor(16 * 256^(l/15)) for l = 0..15 (matches reference float64 computation)
__device__ __constant__ float c_res[LVLS] = {
    16.0f,  23.0f,  33.0f,   48.0f,   70.0f,   101.0f,  147.0f,  212.0f,
    307.0f, 445.0f, 645.0f,  933.0f,  1351.0f, 1955.0f, 2830.0f, 4096.0f
};

__global__ __launch_bounds__(256) void hashgrid_enc_kernel(
    const float* __restrict__ x,       // (N, 3)
    const float* __restrict__ table,   // (T*L, 2)
    float* __restrict__ out,           // (N, L*2)
    unsigned total)                    // N * L
{
    const unsigned g = blockIdx.x * 256u + threadIdx.x;
    if (g >= total) return;

    const unsigned n = g >> 4;    // point index   (16 lanes share one point)
    const unsigned l = g & 15u;   // level index

    // Streaming-in point coords: non-temporal (read once, don't pollute L2,
    // which we want to keep filled with the 64MB hash table). The 16 lanes of
    // a point group issue identical addresses -> coalescer broadcasts.
    const float* xp = x + n * 3u;
    const float x0 = __builtin_nontemporal_load(xp + 0);
    const float x1 = __builtin_nontemporal_load(xp + 1);
    const float x2 = __builtin_nontemporal_load(xp + 2);

    // Prefetch the x stream ~64 points ahead (gfx1250 global_prefetch_b8).
    // Unconditional: all 16 lanes of a point group present the same address,
    // which the coalescer merges — cheaper than an EXEC-mask branch.
    // Speculative prefetch past the end of x is silently dropped.
    __builtin_prefetch(xp + 3u * 64u, 0, 0);

    const float res = c_res[l];

    const float sx = x0 * res;
    const float sy = x1 * res;
    const float sz = x2 * res;
    const float fx = floorf(sx);
    const float fy = floorf(sy);
    const float fz = floorf(sz);
    const float wx = sx - fx;
    const float wy = sy - fy;
    const float wz = sz - fz;
    const int ix = (int)fx;
    const int iy = (int)fy;
    const int iz = (int)fz;

    // hash(ix,iy,iz) = (ix*1 ^ iy*P2 ^ iz*P3) & (T-1)
    const unsigned P2 = 2654435761u;
    const unsigned P3 = 805459861u;
    const unsigned hx0 = (unsigned)ix;
    const unsigned hx1 = (unsigned)(ix + 1);
    const unsigned hy0 = (unsigned)iy * P2;
    const unsigned hy1 = (unsigned)(iy + 1) * P2;
    const unsigned hz0 = (unsigned)iz * P3;
    const unsigned hz1 = (unsigned)(iz + 1) * P3;

    // 32-bit byte offsets into the table (max 64MB - 8 < 2^26): lets the
    // backend use the SADDR(base) + 32-bit VGPR-offset global-load form
    // instead of materializing per-lane 64-bit addresses.
    const unsigned lbase = l << (LOG2_T + 3);  // l * 4MB
    const unsigned o000 = (((hx0 ^ hy0 ^ hz0) & HMASK) << 3) + lbase;
    const unsigned o001 = (((hx0 ^ hy0 ^ hz1) & HMASK) << 3) + lbase;
    const unsigned o010 = (((hx0 ^ hy1 ^ hz0) & HMASK) << 3) + lbase;
    const unsigned o011 = (((hx0 ^ hy1 ^ hz1) & HMASK) << 3) + lbase;
    const unsigned o100 = (((hx1 ^ hy0 ^ hz0) & HMASK) << 3) + lbase;
    const unsigned o101 = (((hx1 ^ hy0 ^ hz1) & HMASK) << 3) + lbase;
    const unsigned o110 = (((hx1 ^ hy1 ^ hz0) & HMASK) << 3) + lbase;
    const unsigned o111 = (((hx1 ^ hy1 ^ hz1) & HMASK) << 3) + lbase;

    // 8 independent b64 gathers from this level's 4MB slice of the table.
    // Issued back-to-back for max memory-level parallelism; this is the hot
    // reuse data, so default (temporal) caching keeps it resident in L2.
    const char* __restrict__ tb = (const char*)table;
    const vf2 f000 = *(const vf2*)(tb + o000);
    const vf2 f001 = *(const vf2*)(tb + o001);
    const vf2 f010 = *(const vf2*)(tb + o010);
    const vf2 f011 = *(const vf2*)(tb + o011);
    const vf2 f100 = *(const vf2*)(tb + o100);
    const vf2 f101 = *(const vf2*)(tb + o101);
    const vf2 f110 = *(const vf2*)(tb + o110);
    const vf2 f111 = *(const vf2*)(tb + o111);

    // Trilinear weights.
    const float ux = 1.0f - wx;
    const float uy = 1.0f - wy;
    const float uz = 1.0f - wz;
    const float c00 = uy * uz;  // (dy=0,dz=0)
    const float c01 = uy * wz;  // (dy=0,dz=1)
    const float c10 = wy * uz;  // (dy=1,dz=0)
    const float c11 = wy * wz;  // (dy=1,dz=1)

    const float w000 = ux * c00;
    const float w001 = ux * c01;
    const float w010 = ux * c10;
    const float w011 = ux * c11;
    const float w100 = wx * c00;
    const float w101 = wx * c01;
    const float w110 = wx * c10;
    const float w111 = wx * c11;

    float a0 = f000.x * w000;
    float a1 = f000.y * w000;
    a0 = fmaf(f001.x, w001, a0);  a1 = fmaf(f001.y, w001, a1);
    a0 = fmaf(f010.x, w010, a0);  a1 = fmaf(f010.y, w010, a1);
    a0 = fmaf(f011.x, w011, a0);  a1 = fmaf(f011.y, w011, a1);
    a0 = fmaf(f100.x, w100, a0);  a1 = fmaf(f100.y, w100, a1);
    a0 = fmaf(f101.x, w101, a0);  a1 = fmaf(f101.y, w101, a1);
    a0 = fmaf(f110.x, w110, a0);  a1 = fmaf(f110.y, w110, a1);
    a0 = fmaf(f111.x, w111, a0);  a1 = fmaf(f111.y, w111, a1);

    // out[n][l*2 + {0,1}] == ((vf2*)out)[g] -> a wave32 writes 256 contiguous
    // bytes. Non-temporal: output is 128MB write-once data; keep it out of L2.
    vf2 r;
    r.x = a0;
    r.y = a1;
    __builtin_nontemporal_store(r, (vf2*)out + g);
}

extern "C" void kernel_launch(void* const* d_in, const int* in_sizes, int n_in,
                              void* d_out, int out_size, void* d_ws, size_t ws_size,
                              hipStream_t stream) {
    const float* x     = (const float*)d_in[0];   // (N, 3) float32
    const float* table = (const float*)d_in[1];   // (T*L, 2) float32
    float* out = (float*)d_out;                   // (N, L*2) float32

    const unsigned npts  = (unsigned)(in_sizes[0] / 3);
    const unsigned total = npts * (unsigned)LVLS; // 16,777,216 threads

    const unsigned block = 256u;
    const unsigned grid  = (total + block - 1u) / block;

    hipLaunchKernelGGL(hashgrid_enc_kernel, dim3(grid), dim3(block), 0, stream,
                       x, table, out, total);
}